// SGDWBackprop_33844342292863
// MI455X (gfx1250) — compile-verified
//
#include <hip/hip_runtime.h>

// ---------------------------------------------------------------------------
// SGDW fused optimizer step for MI455X (gfx1250).
// Memory-bound streaming kernel: 16 B traffic / element @ 23.3 TB/s => ~42 us.
// Data path: CDNA5 async global->LDS b128 loads (ASYNCcnt) + NT b128 stores.
// ---------------------------------------------------------------------------

typedef float v4f __attribute__((ext_vector_type(4)));
typedef int   v4i __attribute__((ext_vector_type(4)));

#define LR   0.01f
#define MOM  0.9f
#define WDC  0.999999f   /* 1 - LR*WD = 1 - 1e-6 */
#define THREADS 256

// ---- CDNA5 async global->LDS 16-byte copy (per-lane, EXEC-predicated) -----
// Builtin signature (from hipcc diagnostic): (v4i AS1*, v4i AS3*, imm, imm)
__device__ __forceinline__ void async_g2l_b128(const void* gptr, void* lds_ptr) {
#if __has_builtin(__builtin_amdgcn_global_load_async_to_lds_b128)
  __builtin_amdgcn_global_load_async_to_lds_b128(
      (__attribute__((address_space(1))) v4i*)(v4i*)gptr,
      (__attribute__((address_space(3))) v4i*)(v4i*)lds_ptr,
      /*imm offset*/ 0, /*cpol*/ 0);
#elif defined(__AMDGCN__)
  unsigned lds_off =
      (unsigned)(unsigned long long)(__attribute__((address_space(3))) void*)lds_ptr;
  asm volatile("global_load_async_to_lds_b128 %0, %1, off"
               :: "v"(lds_off), "v"(gptr)
               : "memory");
#else
  (void)gptr; (void)lds_ptr;
#endif
}

template <int N>
__device__ __forceinline__ void wait_async() {
#if __has_builtin(__builtin_amdgcn_s_wait_asynccnt)
  __builtin_amdgcn_s_wait_asynccnt(N);
#elif defined(__AMDGCN__)
  asm volatile("s_wait_asynccnt %0" :: "n"(N) : "memory");
#endif
  // keep the compiler from hoisting LDS reads above the wait
  __atomic_signal_fence(__ATOMIC_ACQ_REL);
}

__device__ __forceinline__ v4f sgdw4(v4f P, v4f G, v4f V) {
  v4f vel = MOM * V + G;          // velocity = momentum*v + grad
  return P * WDC - LR * vel;      // decoupled WD then SGD step
}

__global__ __launch_bounds__(THREADS) void sgdw_async_kernel(
    const float* __restrict__ p, const float* __restrict__ g,
    const float* __restrict__ v, float* __restrict__ o, long long n4) {
  // 2 pipeline stages x {p,g,v} x 256 lanes x 16 B = 24 KB LDS
  __shared__ v4f sb[2][3][THREADS];

  const int tid = threadIdx.x;
  const long long i0 = (long long)blockIdx.x * (2 * THREADS) + tid;
  const long long i1 = i0 + THREADS;

  const v4f* p4 = (const v4f*)p;
  const v4f* g4 = (const v4f*)g;
  const v4f* v4 = (const v4f*)v;
  v4f*       o4 = (v4f*)o;

  // Issue both stages' async loads up front (6 x b128 in flight per wave,
  // zero VGPR destinations -> deep MLP at full occupancy).
  if (i0 < n4) {
    async_g2l_b128(p4 + i0, &sb[0][0][tid]);
    async_g2l_b128(g4 + i0, &sb[0][1][tid]);
    async_g2l_b128(v4 + i0, &sb[0][2][tid]);
  }
  if (i1 < n4) {
    async_g2l_b128(p4 + i1, &sb[1][0][tid]);
    async_g2l_b128(g4 + i1, &sb[1][1][tid]);
    async_g2l_b128(v4 + i1, &sb[1][2][tid]);
  }

  // Stage 0 ready once at most the 3 stage-1 loads remain outstanding.
  // (Under wave divergence both waits may execute; wait<0> is then merely
  // stricter, still correct.)
  if (i1 < n4) {
    wait_async<3>();
  } else {
    wait_async<0>();
  }

  if (i0 < n4) {
    v4f r = sgdw4(sb[0][0][tid], sb[0][1][tid], sb[0][2][tid]);
    __builtin_nontemporal_store(r, o4 + i0);   // streaming store, no L2 reuse
  }

  wait_async<0>();   // stage 1 complete

  if (i1 < n4) {
    v4f r = sgdw4(sb[1][0][tid], sb[1][1][tid], sb[1][2][tid]);
    __builtin_nontemporal_store(r, o4 + i1);
  }
}

extern "C" void kernel_launch(void* const* d_in, const int* in_sizes, int n_in,
                              void* d_out, int out_size, void* d_ws, size_t ws_size,
                              hipStream_t stream) {
  (void)n_in; (void)out_size; (void)d_ws; (void)ws_size;
  // setup_inputs order: p0,g0,v0, p1,g1,v1, p2,g2,v2, p3,g3,v3
  long long off = 0;
  for (int t = 0; t < 4; ++t) {
    const int base = 3 * t;
    const long long n  = in_sizes[base];     // elements (all multiples of 1024)
    const long long n4 = n >> 2;             // float4 chunks
    const int blocks = (int)((n4 + 2 * THREADS - 1) / (2 * THREADS));
    sgdw_async_kernel<<<blocks, THREADS, 0, stream>>>(
        (const float*)d_in[base + 0],
        (const float*)d_in[base + 1],
        (const float*)d_in[base + 2],
        (float*)d_out + off, n4);
    off += n;
  }
}